// InterpretableMultiHeadAttention_28415503630800
// MI455X (gfx1250) — compile-verified
//
#include <hip/hip_runtime.h>

#define Bv 32
#define Sv 512
#define Dv 256
#define Hv 8
#define DHv 32

typedef __attribute__((ext_vector_type(2))) float v2f;
typedef __attribute__((ext_vector_type(4))) float v4f;
typedef __attribute__((ext_vector_type(8))) float v8f;

// ---------------------------------------------------------------------------
// WMMA f32 GEMM: C[M, 256] = A[M, K] * B (+ bias), optionally batched.
//  WT = true : B given as W[N x K] row-major (compute A * W^T)   -- projections
//  WT = false: B given as  [K x N] row-major (compute A * B)     -- avg * V
// Block = 128 threads (4 waves). Each block: 16 rows x 256 cols.
// Each wave: 16 x 64 strip = 4 x v8f accumulators.
// A tile (16 x 128 per chunk) is DMA'd global->LDS with
// GLOBAL_LOAD_ASYNC_TO_LDS_B128 (ASYNCcnt), double-buffered so the copy of
// chunk i+1 overlaps the 128 WMMAs of chunk i.
// VGPR layouts per CDNA5 ISA 7.12.2:
//  A 16x4 f32 : v[0] = (M=lane&15, K=2*(lane>>4)),   v[1] = K+1
//  B 4x16 f32 : v[0] = (K=2*(lane>>4), N=lane&15),   v[1] = K+1
//  C 16x16    : vgpr j = (M = j + 8*(lane>>4), N = lane&15)
// ---------------------------------------------------------------------------
template <bool WT>
__global__ __launch_bounds__(128) void gemm_wmma_f32(
    const float* __restrict__ A, const float* __restrict__ Bm,
    const float* __restrict__ bias, float* __restrict__ C,
    int K, int lda, int ldb, int ldc, long sA, long sB, long sC) {
  __shared__ float As[2][16 * 128];  // 2 x 8 KB ping-pong

  const int tid  = threadIdx.x;
  const int wave = tid >> 5;
  const int lane = tid & 31;
  const int row  = lane & 15;
  const int kh   = lane >> 4;  // 0/1: selects K pair within fragment
  const int m0   = blockIdx.x * 16;
  const int n0   = wave * 64;

  const float* Ab = A + (long)blockIdx.z * sA;
  const float* Bb = Bm + (long)blockIdx.z * sB;
  float* Cb       = C + (long)blockIdx.z * sC;

  // Async DMA of one 16x128 A chunk into As[ib]: 512 b128 slots / 128 threads.
  auto issue_chunk = [&](int ib, int kc) {
#pragma unroll
    for (int i = 0; i < 4; ++i) {
      const int slot = i * 128 + tid;  // 0..511
      const int r    = slot >> 5;      // row 0..15
      const int c4   = slot & 31;      // float4 column
      const unsigned dst =
          (unsigned)(uintptr_t)(&As[ib][r * 128 + c4 * 4]);
      const float* src = Ab + (long)(m0 + r) * lda + kc + c4 * 4;
      asm volatile("global_load_async_to_lds_b128 %0, %1, off"
                   :: "v"(dst), "v"(src) : "memory");
    }
  };

  v8f acc[4] = {v8f{}, v8f{}, v8f{}, v8f{}};

  issue_chunk(0, 0);
  int ib = 0;
  for (int kc = 0; kc < K; kc += 128, ib ^= 1) {
    const bool more = (kc + 128 < K);
    if (more) {
      issue_chunk(ib ^ 1, kc + 128);  // overlap next copy with this compute
      asm volatile("s_wait_asynccnt 0x4" ::: "memory");
    } else {
      asm volatile("s_wait_asynccnt 0x0" ::: "memory");
    }
    __syncthreads();  // all waves' portions of As[ib] landed

    const float* Ac = &As[ib][0];
#pragma unroll 4
    for (int kk = 0; kk < 128; kk += 4) {
      const int kb = kk + 2 * kh;
      v2f a;
      a.x = Ac[row * 128 + kb];
      a.y = Ac[row * 128 + kb + 1];

      v2f bf[4];
      if (WT) {
#pragma unroll
        for (int nt = 0; nt < 4; ++nt) {
          const float* p = Bb + (long)(n0 + nt * 16 + row) * ldb + kc + kb;
          bf[nt].x = p[0];
          bf[nt].y = p[1];
        }
      } else {
        const float* p = Bb + (long)(kc + kb) * ldb + n0 + row;
        const float* q = p + ldb;
#pragma unroll
        for (int nt = 0; nt < 4; ++nt) {
          bf[nt].x = p[nt * 16];
          bf[nt].y = q[nt * 16];
        }
      }
#pragma unroll
      for (int nt = 0; nt < 4; ++nt) {
        acc[nt] = __builtin_amdgcn_wmma_f32_16x16x4_f32(
            false, a, false, bf[nt], (short)0, acc[nt], false, false);
      }
    }
    __syncthreads();  // all reads of As[ib] done -> next iter may overwrite
  }

#pragma unroll
  for (int nt = 0; nt < 4; ++nt) {
    const int c = n0 + nt * 16 + row;
    const float bb = bias ? bias[c] : 0.0f;
#pragma unroll
    for (int j = 0; j < 8; ++j) {
      const int m = m0 + j + 8 * kh;
      Cb[(long)m * ldc + c] = acc[nt][j] + bb;
    }
  }
}

// ---------------------------------------------------------------------------
// Scores + sparsemax + head-mean, fused. One 512-thread block per (b, s).
// For each head: score row (q[b,s,h,:] . k[b,t,h,:] / sqrt(32), diag -> -inf),
// bitonic descending sort in LDS, inclusive scan, support count via atomicMax,
// tau threshold, accumulate attn/H into the avg row. Writes avg_attention.
// ---------------------------------------------------------------------------
__global__ __launch_bounds__(512) void sparsemax_attn(
    const float* __restrict__ q, const float* __restrict__ k,
    float* __restrict__ avg_out) {
  __shared__ float qrow[Dv];
  __shared__ float sc[Sv];
  __shared__ float srt[Sv];
  __shared__ float cs[Sv];
  __shared__ float avg[Sv];
  __shared__ int ksel;

  const int t = threadIdx.x;
  const int s = blockIdx.x;
  const int b = blockIdx.y;

  if (t < Dv) qrow[t] = q[((long)(b * Sv + s)) * Dv + t];
  avg[t] = 0.0f;
  __syncthreads();

  const float rs = 0.17677669529663687f;  // 1/sqrt(32)

  for (int h = 0; h < Hv; ++h) {
    // score[t] = q[b,s,h,:] . k[b,t,h,:]
    const float* kp = k + ((long)(b * Sv + t)) * Dv + h * DHv;
    const float* qp = qrow + h * DHv;
    float acc = 0.0f;
#pragma unroll
    for (int e = 0; e < DHv; e += 4) {
      v4f kv = *(const v4f*)(kp + e);
      acc = fmaf(qp[e + 0], kv.x, acc);
      acc = fmaf(qp[e + 1], kv.y, acc);
      acc = fmaf(qp[e + 2], kv.z, acc);
      acc = fmaf(qp[e + 3], kv.w, acc);
    }
    acc *= rs;
    if (t == s) acc = -__builtin_inff();
    sc[t]  = acc;
    srt[t] = acc;
    if (t == 0) ksel = 0;
    __syncthreads();

    // Bitonic sort, descending.
    for (int ksz = 2; ksz <= Sv; ksz <<= 1) {
      for (int j = ksz >> 1; j > 0; j >>= 1) {
        const int ixj = t ^ j;
        if (ixj > t) {
          const float a0 = srt[t];
          const float a1 = srt[ixj];
          const bool desc = ((t & ksz) == 0);
          if (desc ? (a0 < a1) : (a0 > a1)) {
            srt[t]   = a1;
            srt[ixj] = a0;
          }
        }
        __syncthreads();
      }
    }

    // Inclusive scan (Hillis-Steele).
    cs[t] = srt[t];
    __syncthreads();
    for (int off = 1; off < Sv; off <<= 1) {
      const float v = (t >= off) ? cs[t - off] : 0.0f;
      __syncthreads();
      if (t >= off) cs[t] += v;
      __syncthreads();
    }

    // Support: 1 + k*zs > cumsum  (strict; -inf rows -> false).
    const float zs = srt[t];
    const bool sup = (1.0f + (float)(t + 1) * zs) > cs[t];
    if (sup) atomicMax(&ksel, t + 1);
    __syncthreads();

    const int kn    = ksel;
    const float tau = (cs[kn - 1] - 1.0f) / (float)kn;
    avg[t] += fmaxf(sc[t] - tau, 0.0f) * (1.0f / (float)Hv);
    __syncthreads();
  }

  avg_out[((long)(b * Sv + s)) * Sv + t] = avg[t];
}

// ---------------------------------------------------------------------------
extern "C" void kernel_launch(void* const* d_in, const int* in_sizes, int n_in,
                              void* d_out, int out_size, void* d_ws,
                              size_t ws_size, hipStream_t stream) {
  (void)in_sizes; (void)n_in; (void)out_size; (void)ws_size;

  const float* x  = (const float*)d_in[0];
  const float* Wq = (const float*)d_in[1];  // [H,DH,D] == [256,256] flat
  const float* bq = (const float*)d_in[2];  // [256] flat
  const float* Wk = (const float*)d_in[3];
  const float* bk = (const float*)d_in[4];
  const float* Wv = (const float*)d_in[5];  // [D,D]
  const float* bv = (const float*)d_in[6];
  const float* Wo = (const float*)d_in[7];
  const float* bo = (const float*)d_in[8];

  const long BSD = (long)Bv * Sv * Dv;  // 4,194,304
  float* outp = (float*)d_out;          // [B,S,D]
  float* avgp = outp + BSD;             // [B,S,S]

  float* qws = (float*)d_ws;            // q as [B,S,H*DH]
  float* kws = qws + BSD;               // k as [B,S,H*DH]
  float* vws = kws + BSD;               // v  [B,S,D]
  float* tws = vws + BSD;               // avg*v [B,S,D]

  const dim3 blk(128);
  const dim3 gProj(Bv * Sv / 16, 1, 1);

  // Q/K/V projections: [16384 x 256] = x * W^T + bias
  gemm_wmma_f32<true><<<gProj, blk, 0, stream>>>(x, Wq, bq, qws, Dv, Dv, Dv, Dv, 0, 0, 0);
  gemm_wmma_f32<true><<<gProj, blk, 0, stream>>>(x, Wk, bk, kws, Dv, Dv, Dv, Dv, 0, 0, 0);
  gemm_wmma_f32<true><<<gProj, blk, 0, stream>>>(x, Wv, bv, vws, Dv, Dv, Dv, Dv, 0, 0, 0);

  // Fused scores + sparsemax + head-mean -> avg_attention (second output)
  sparsemax_attn<<<dim3(Sv, Bv), 512, 0, stream>>>(qws, kws, avgp);

  // tmp[b] = avg[b] * v[b]   (batched, B non-transposed)
  gemm_wmma_f32<false><<<dim3(Sv / 16, 1, Bv), blk, 0, stream>>>(
      avgp, vws, nullptr, tws, Sv, Sv, Dv, Dv, (long)Sv * Sv, (long)Sv * Dv,
      (long)Sv * Dv);

  // out = tmp * Wo^T + bo
  gemm_wmma_f32<true><<<gProj, blk, 0, stream>>>(tws, Wo, bo, outp, Dv, Dv, Dv, Dv, 0, 0, 0);
}